// RoIPooling_18313740550509
// MI455X (gfx1250) — compile-verified
//
#include <hip/hip_runtime.h>
#include <stdint.h>

typedef float v4f __attribute__((ext_vector_type(4)));

#define CROP  7
#define IMG_H 64
#define IMG_W 64
#define NCH   256
#define QUADS 64          // 256 channels / 4 per lane

// One block (64 threads = 2 wave32) produces one (box, oy) row: 7 pixels x 256 ch.
// Double-buffered async global->LDS pipeline stages the 4 bilinear corners of the
// next pixel (4 x 1KB) while the current pixel is blended and NT-streamed out.
__global__ __launch_bounds__(64, 8)
void crop_resize_async_kernel(const float* __restrict__ feats,
                              const float* __restrict__ boxes,
                              const int*   __restrict__ box_ind,
                              float*       __restrict__ out)
{
    __shared__ v4f smem[2][4][QUADS];   // 2 stages * 4 corners * 64 quads * 16B = 8 KB

    const int tid = threadIdx.x;        // 0..63 -> channel quad
    const int row = blockIdx.x;         // box*7 + oy
    const int box = row / CROP;
    const int oy  = row - box * CROP;

    // Block-uniform box parameters (compiler scalarizes to SALU/SMEM).
    const float y1 = boxes[box * 4 + 0];
    const float x1 = boxes[box * 4 + 1];
    const float y2 = boxes[box * 4 + 2];
    const float x2 = boxes[box * 4 + 3];
    const int   bi = box_ind[box];

    // y sample for this output row (reference: lo*(H-1) + i*step)
    const float ystep = (y2 - y1) * ((float)(IMG_H - 1) / (float)(CROP - 1));
    const float ys    = y1 * (float)(IMG_H - 1) + (float)oy * ystep;
    const float y0f   = floorf(ys);
    const float wy    = ys - y0f;              // unclipped fractional weight
    int y0i = (int)y0f;
    y0i = y0i < 0 ? 0 : (y0i > IMG_H - 1 ? IMG_H - 1 : y0i);
    int y1i = y0i + 1;
    y1i = y1i > IMG_H - 1 ? IMG_H - 1 : y1i;
    const bool vy = (ys >= 0.0f) && (ys <= (float)(IMG_H - 1));

    // Precompute all 7 x samples (fully unrolled, uniform per block).
    const float xstep = (x2 - x1) * ((float)(IMG_W - 1) / (float)(CROP - 1));
    int   xi0[CROP], xi1[CROP];
    float wxv[CROP];
    bool  vxv[CROP];
    #pragma unroll
    for (int p = 0; p < CROP; ++p) {
        const float xs  = x1 * (float)(IMG_W - 1) + (float)p * xstep;
        const float x0f = floorf(xs);
        wxv[p] = xs - x0f;
        int x0 = (int)x0f;
        x0 = x0 < 0 ? 0 : (x0 > IMG_W - 1 ? IMG_W - 1 : x0);
        int xb = x0 + 1;
        xb = xb > IMG_W - 1 ? IMG_W - 1 : xb;
        xi0[p] = x0;
        xi1[p] = xb;
        vxv[p] = (xs >= 0.0f) && (xs <= (float)(IMG_W - 1));
    }

    const uint64_t imgBase = (uint64_t)bi * (uint64_t)(IMG_H * IMG_W * NCH);

    auto corner_ptr = [&](int y, int x) -> const v4f* {
        return (const v4f*)(feats + imgBase + ((uint64_t)y * IMG_W + (uint64_t)x) * NCH) + tid;
    };

    // Stage pixel p's 4 corner channel-quads into LDS via async DMA (ASYNCcnt path).
    auto issue_stage = [&](int p, int stage) {
        const uint64_t a00 = (uint64_t)(uintptr_t)corner_ptr(y0i, xi0[p]);
        const uint64_t a01 = (uint64_t)(uintptr_t)corner_ptr(y0i, xi1[p]);
        const uint64_t a10 = (uint64_t)(uintptr_t)corner_ptr(y1i, xi0[p]);
        const uint64_t a11 = (uint64_t)(uintptr_t)corner_ptr(y1i, xi1[p]);
        const uint32_t l00 = (uint32_t)(uintptr_t)&smem[stage][0][tid];
        const uint32_t l01 = (uint32_t)(uintptr_t)&smem[stage][1][tid];
        const uint32_t l10 = (uint32_t)(uintptr_t)&smem[stage][2][tid];
        const uint32_t l11 = (uint32_t)(uintptr_t)&smem[stage][3][tid];
        asm volatile("global_load_async_to_lds_b128 %0, %1, off" :: "v"(l00), "v"(a00) : "memory");
        asm volatile("global_load_async_to_lds_b128 %0, %1, off" :: "v"(l01), "v"(a01) : "memory");
        asm volatile("global_load_async_to_lds_b128 %0, %1, off" :: "v"(l10), "v"(a10) : "memory");
        asm volatile("global_load_async_to_lds_b128 %0, %1, off" :: "v"(l11), "v"(a11) : "memory");
    };

    issue_stage(0, 0);

    float* outRow = out + (uint64_t)row * (uint64_t)(CROP * NCH);

    #pragma unroll
    for (int p = 0; p < CROP; ++p) {
        const int stage = p & 1;
        if (p + 1 < CROP) {
            issue_stage(p + 1, (p + 1) & 1);
            // 8 async ops in flight; loads retire in order -> oldest 4 (stage p) done.
            asm volatile("s_wait_asynccnt 0x4" ::: "memory");
        } else {
            asm volatile("s_wait_asynccnt 0x0" ::: "memory");
        }

        // Each wave reads only the LDS data it staged itself -> no barrier needed.
        v4f c00 = smem[stage][0][tid];
        v4f c01 = smem[stage][1][tid];
        v4f c10 = smem[stage][2][tid];
        v4f c11 = smem[stage][3][tid];

        const float wx = wxv[p];
        v4f top = c00 + (c01 - c00) * wx;
        v4f bot = c10 + (c11 - c10) * wx;
        v4f res = top + (bot - top) * wy;

        if (!(vy && vxv[p])) {
            v4f zero = {0.0f, 0.0f, 0.0f, 0.0f};
            res = zero;
        }

        // 200 MB output stream: non-temporal so feats (33.5 MB) stays L2-resident.
        __builtin_nontemporal_store(res, (v4f*)(outRow + (uint64_t)p * NCH) + tid);
    }
}

extern "C" void kernel_launch(void* const* d_in, const int* in_sizes, int n_in,
                              void* d_out, int out_size, void* d_ws, size_t ws_size,
                              hipStream_t stream) {
    const float* feats   = (const float*)d_in[0];
    const float* boxes   = (const float*)d_in[1];
    const int*   box_ind = (const int*)d_in[2];
    float*       out     = (float*)d_out;

    const int nboxes = in_sizes[1] / 4;      // 4000
    dim3 grid(nboxes * CROP);                // one block per (box, oy) row
    dim3 block(64);
    crop_resize_async_kernel<<<grid, block, 0, stream>>>(feats, boxes, box_ind, out);
}